// IL_NIQE_86320252715842
// MI455X (gfx1250) — compile-verified
//
#include <hip/hip_runtime.h>
#include <hip/hip_bf16.h>

// ---------------- CDNA5 feature detection ----------------
#if defined(__has_builtin)
#if __has_builtin(__builtin_amdgcn_global_load_async_to_lds_b128)
#define HAS_ASYNC_LDS 1
#endif
#if __has_builtin(__builtin_amdgcn_s_wait_asynccnt)
#define HAS_WAIT_ASYNC 1
#endif
#if __has_builtin(__builtin_amdgcn_wmma_f32_16x16x4_f32)
#define HAS_WMMA_F32X4 1
#endif
#endif

typedef float v2f __attribute__((ext_vector_type(2)));
typedef float v8f __attribute__((ext_vector_type(8)));
typedef int   v4i __attribute__((ext_vector_type(4)));

#define NPX   (84 * 84)     // 7056 elements per block
#define NF4   (84 * 21)     // 1764 float4 per block
#define PSTR  33            // padded LDS stride for partials (>=32, spreads banks)
#define NTAB  9801

// ---------------- r(gamma) lookup table ----------------
__global__ void rgam_kernel(float* __restrict__ rgam) {
    int i = blockIdx.x * blockDim.x + threadIdx.x;
    if (i < NTAB) {
        float g = 0.2f + 0.001f * (float)i;
        rgam[i] = expf(2.0f * lgammaf(2.0f / g) - lgammaf(1.0f / g) - lgammaf(3.0f / g));
    }
}

// accumulate one value into the 5 AGGD sums
__device__ inline void acc5(float v, float* a) {
    float v2 = v * v;
    bool neg = v < 0.0f, pos = v > 0.0f;
    a[0] += neg ? 1.0f : 0.0f;   // cnt_l
    a[1] += pos ? 1.0f : 0.0f;   // cnt_r
    a[2] += neg ? v2 : 0.0f;     // sumsq_l
    a[3] += pos ? v2 : 0.0f;     // sumsq_r
    a[4] += fabsf(v);            // sum |v|
}

__device__ inline void async_copy16(const float* g, float* l) {
#if defined(HAS_ASYNC_LDS)
    typedef __attribute__((address_space(1))) v4i g4_t;   // global v4i*
    typedef __attribute__((address_space(3))) v4i l4_t;   // LDS v4i*
    __builtin_amdgcn_global_load_async_to_lds_b128((g4_t*)(float*)g, (l4_t*)l, 0, 0);
#else
    *(float4*)l = *(const float4*)g;
#endif
}

__global__ __launch_bounds__(256) void niqe_kernel(const float* __restrict__ x,
                                                   const float* __restrict__ rgam,
                                                   float* __restrict__ out) {
    __shared__ __align__(16) float blk[NPX];         // 28224 B
    __shared__ float part[256 * PSTR];               // 33792 B
    __shared__ float totals[32];

    const int tid  = threadIdx.x;
    const int lane = tid & 31;
    const int wid  = tid >> 5;
    const int bid  = blockIdx.x;
    const int img  = bid / 36;
    const int bi   = bid % 36;
    const int bh   = bi / 6;
    const int bw   = bi % 6;

    const float* base = x + (size_t)img * (504 * 504) + (size_t)(bh * 84) * 504 + (size_t)(bw * 84);

    // ---- stage 1: async copy 84x84 tile into LDS (16B-aligned b128 transfers) ----
    for (int j = tid; j < NF4; j += 256) {
        int r  = j / 21;
        int c4 = j - r * 21;
        async_copy16(base + r * 504 + c4 * 4, &blk[r * 84 + c4 * 4]);
    }
#if defined(HAS_ASYNC_LDS)
#if defined(HAS_WAIT_ASYNC)
    __builtin_amdgcn_s_wait_asynccnt(0);
#else
    asm volatile("s_wait_asynccnt 0" ::: "memory");
#endif
#endif
    __syncthreads();

    // ---- stage 2: per-thread accumulation of 25 sums (5 variants x 5 quantities) ----
    float a[25];
#pragma unroll
    for (int v = 0; v < 25; ++v) a[v] = 0.0f;

    for (int i = tid; i < NPX; i += 256) {
        int r = i / 84;
        int c = i - r * 84;
        int rm1 = (r == 0) ? 83 : r - 1;
        int cm1 = (c == 0) ? 83 : c - 1;
        int cp1 = (c == 83) ? 0 : c + 1;
        float e = blk[r * 84 + c];
        acc5(e, a + 0);                              // raw block
        acc5(e * blk[r * 84 + cm1],   a + 5);        // shift (0,1)
        acc5(e * blk[rm1 * 84 + c],   a + 10);       // shift (1,0)
        acc5(e * blk[rm1 * 84 + cm1], a + 15);       // shift (1,1)
        acc5(e * blk[rm1 * 84 + cp1], a + 20);       // shift (1,-1)
    }
#pragma unroll
    for (int v = 0; v < 25; ++v) part[tid * PSTR + v] = a[v];
#pragma unroll
    for (int v = 25; v < 32; ++v) part[tid * PSTR + v] = 0.0f;  // keep junk out of WMMA cols
    __syncthreads();

    // ---- stage 3: reduce part[256][25] across threads via V_WMMA_F32_16X16X4_F32 ----
    // D = sum_K ones(16x4) x B(4x16), C accumulating: every D row == column sums.
    // Summing over ALL of K makes the result independent of the exact K->(vgpr,half) map.
#if defined(HAS_WMMA_F32X4)
    if (wid < 2) {                                   // wave-uniform branch, EXEC all ones
        const int n    = lane & 15;
        const int half = lane >> 4;
        const int col  = wid * 16 + n;               // quantity index 0..31
        v2f ones; ones.x = 1.0f; ones.y = 1.0f;
        v8f acc = {0.f, 0.f, 0.f, 0.f, 0.f, 0.f, 0.f, 0.f};
        for (int t0 = 0; t0 < 256; t0 += 4) {
            v2f b;
            b.x = part[(t0 + half)     * PSTR + col];
            b.y = part[(t0 + 2 + half) * PSTR + col];
            acc = __builtin_amdgcn_wmma_f32_16x16x4_f32(
                false, ones, false, b, (short)0, acc, false, false);
        }
        if (half == 0 && col < 25) totals[col] = acc[0];  // D[0][n] = colsum
    }
#else
    if (wid == 0) {
        for (int v = lane; v < 25; v += 32) {
            float s = 0.0f;
            for (int t = 0; t < 256; ++t) s += part[t * PSTR + v];
            totals[v] = s;
        }
    }
#endif
    __syncthreads();

    // ---- stage 4: one wave per AGGD variant: grid-search argmin + lgamma tail ----
    if (wid < 5) {
        const int w = wid;
        float cl   = totals[w * 5 + 0];
        float cr   = totals[w * 5 + 1];
        float ssl  = totals[w * 5 + 2];
        float ssr  = totals[w * 5 + 3];
        float sabs = totals[w * 5 + 4];

        float lstd = sqrtf(ssl / cl);
        float rstd = sqrtf(ssr / cr);
        float gh   = lstd / rstd;
        const float N = (float)NPX;
        float rhat = (sabs * sabs) / (N * (ssl + ssr));  // mean|x|^2 / mean(x^2)
        float g2   = gh * gh;
        float rhatnorm = rhat * (g2 * gh + 1.0f) * (gh + 1.0f) / ((g2 + 1.0f) * (g2 + 1.0f));

        float bestd = 3.4e38f;
        int   besti = 0;
        for (int i = lane; i < NTAB; i += 32) {
            float d = rgam[i] - rhatnorm;
            d *= d;
            if (d < bestd) { bestd = d; besti = i; }   // first-min within lane stream
        }
        // cross-lane argmin, prefer smaller index on ties (matches jnp.argmin)
        for (int off = 16; off > 0; off >>= 1) {
            float od = __shfl_xor(bestd, off, 32);
            int   oi = __shfl_xor(besti, off, 32);
            if (od < bestd || (od == bestd && oi < besti)) { bestd = od; besti = oi; }
        }

        if (lane == 0) {
            float alpha = 0.2f + 0.001f * (float)besti;
            float scale = sqrtf(expf(lgammaf(1.0f / alpha) - lgammaf(3.0f / alpha)));
            float bl = lstd * scale;
            float br = rstd * scale;
            float* o = out + (size_t)bid * 18;
            if (w == 0) {
                o[0] = alpha;
                o[1] = 0.5f * (bl + br);
            } else {
                int fb = 2 + (w - 1) * 4;
                o[fb + 0] = alpha;
                o[fb + 1] = (br - bl) * expf(lgammaf(2.0f / alpha) - lgammaf(1.0f / alpha));
                o[fb + 2] = bl;
                o[fb + 3] = br;
            }
        }
    }
}

extern "C" void kernel_launch(void* const* d_in, const int* in_sizes, int n_in,
                              void* d_out, int out_size, void* d_ws, size_t ws_size,
                              hipStream_t stream) {
    const float* x   = (const float*)d_in[0];
    float*       out = (float*)d_out;
    float*       rg  = (float*)d_ws;          // 9801 floats of scratch

    const int b    = in_sizes[0] / (504 * 504);  // 64 images
    const int nblk = b * 36;

    rgam_kernel<<<(NTAB + 255) / 256, 256, 0, stream>>>(rg);
    niqe_kernel<<<nblk, 256, 0, stream>>>(x, rg, out);
}